// CCNN_47742856462743
// MI455X (gfx1250) — compile-verified
//
#include <hip/hip_runtime.h>

#define BS 8
#define L0 2048
#define L1 2049
#define CCH 32
#define HD 16
#define K5 5
#define BSL (BS * L1)          // 16392 rows
#define TM 16                  // rows per block tile
#define KTOT 576               // 512 (outer) + 32 (bias path) + 32 (skip path)
#define AST 577                // padded LDS stride for A2 (odd -> bank friendly)
#define NEG 0.1f
#define BN_EPS 1e-5f

typedef float v2f __attribute__((ext_vector_type(2)));
typedef float v8f __attribute__((ext_vector_type(8)));

__device__ __forceinline__ float leaky(float x) { return x >= 0.f ? x : NEG * x; }

// ---------------- workspace layout (float units) ----------------
#define OFF_TIMES   0                         // 16392 (rounded to 16512)
#define OFF_ENCA    16512                     // 524288
#define OFF_ENCB    (OFF_ENCA + 524288)
#define OFF_OUTPRE  (OFF_ENCB + 524288)
#define OFF_W       (OFF_OUTPRE + 524288)     // 4*576*32 = 73728
#define OFF_STATS   (OFF_W + 73728)           // 4 layers * (32 sum + 32 sumsq)
#define OFF_BOS     (OFF_STATS + 256)         // 1 int

// ---------------- init: zero BN stats + bos ----------------
__global__ void k_init(float* stats, int* bos) {
    int t = threadIdx.x;
    if (t < 256) stats[t] = 0.f;
    if (t == 0) *bos = 0;
}

// ---------------- max over event_types ----------------
__global__ void k_maxtype(const int* __restrict__ types, int* bos) {
    int i = blockIdx.x * blockDim.x + threadIdx.x;
    if (i < BS * L0) atomicMax(bos, types[i]);
}

// ---------------- build padded times ----------------
__global__ void k_times(const float* __restrict__ ev_times, float* __restrict__ times) {
    int i = blockIdx.x * blockDim.x + threadIdx.x;
    if (i >= BSL) return;
    int b = i / L1, l = i % L1;
    times[i] = (l == 0) ? 0.f : ev_times[b * L0 + (l - 1)];
}

// ---------------- embedding gather (emb[0] treated as 0; BOS = max+1) ----------------
__global__ void k_embed(const int* __restrict__ types, const float* __restrict__ emb,
                        const int* __restrict__ bos, float* __restrict__ enc) {
    int i = blockIdx.x * blockDim.x + threadIdx.x;     // over BSL*CCH = 524544
    int r = i >> 5, c = i & 31;
    int b = r / L1, l = r % L1;
    int tt = (l == 0) ? (bos[0] + 1) : types[b * L0 + (l - 1)];
    enc[i] = (tt == 0) ? 0.f : emb[tt * CCH + c];
}

// ---------------- build stacked GEMM weights, K-pair interleaved ----------------
// Logical W (576 x 32):
//   rows 0..511:  W[c*16+h][d] = k3W[layer][h][c*32+d]
//   rows 512..543: k3b[layer][c*32+d]
//   rows 544..575: skipW[layer][c][d]
// Stored as float2 pairs: Wst[layer][ (e/2)*32 + d ] = { W[e_even][d], W[e_even+1][d] }
// so a WMMA B operand (lane=col, K pair) is one aligned b64 load.
__global__ void k_wprep(const float* __restrict__ k3W, const float* __restrict__ k3b,
                        const float* __restrict__ skipW, float* __restrict__ Wst) {
    int i = blockIdx.x * blockDim.x + threadIdx.x;     // 4*576*32 = 73728
    if (i >= 4 * KTOT * CCH) return;
    int layer = i / (KTOT * CCH);
    int rem = i % (KTOT * CCH);
    int e = rem / CCH, d = rem % CCH;
    float v;
    if (e < 512) {
        int c = e >> 4, h = e & 15;
        v = k3W[(layer * HD + h) * 1024 + c * 32 + d];
    } else if (e < 544) {
        int c = e - 512;
        v = k3b[layer * 1024 + c * 32 + d];
    } else {
        int c = e - 544;
        v = skipW[(layer * CCH + c) * CCH + d];
    }
    // K-pair interleaved destination
    Wst[layer * KTOT * CCH + (((e >> 1) * CCH + d) << 1) + (e & 1)] = v;
}

// ---------------- fused layer: gather + kernel-MLP + WMMA GEMM + BN stats ----------------
__global__ __launch_bounds__(256)
void k_layer(const float* __restrict__ times, const float* __restrict__ encIn,
             const float* __restrict__ W, const float* __restrict__ k1W,
             const float* __restrict__ k1b, const float* __restrict__ k2W,
             const float* __restrict__ k2b, const float* __restrict__ skipb,
             float* __restrict__ outPre, float* __restrict__ stats, int dil) {
    __shared__ float A2[TM][AST];           // 16 x 577 fused-GEMM A tile
    __shared__ float G[TM][K5][CCH];        // gathered features
    __shared__ float dtv[TM][K5];           // masked time deltas
    __shared__ float mv[TM][K5];            // masks (0/1)
    __shared__ float part[8][TM][CCH];      // per-wave partial C tiles
    __shared__ float sk1W[HD], sk1b[HD], sk2b[HD], sk2W[HD * HD];

    int tile = blockIdx.x;
    int tid = threadIdx.x;
    int lane = tid & 31;
    int wv = tid >> 5;

    // --- stage small MLP weights into LDS ---
    if (tid < HD * HD) sk2W[tid] = k2W[tid];
    if (tid < HD) { sk1W[tid] = k1W[tid]; sk1b[tid] = k1b[tid]; sk2b[tid] = k2b[tid]; }

    // --- phase 0a: dt & mask per (row, tap) ---
    if (tid < TM * K5) {
        int t = tid / K5, k = tid % K5;
        int r = tile * TM + t;
        int rr = (r < BSL) ? r : 0;
        int b = rr / L1, l = rr % L1;
        int idx = l - k * dil;
        int ic = idx < 0 ? 0 : idx;
        float tl = times[b * L1 + l];
        float ti = times[b * L1 + ic];
        float m = (idx >= 0 && ti != 0.f && tl != 0.f && r < BSL) ? 1.f : 0.f;
        mv[t][k] = m;
        dtv[t][k] = (tl - ti) * m;
    }
    // --- phase 0b: gather features (clipped indices, unmasked) ---
    for (int e = tid; e < TM * K5 * CCH; e += 256) {
        int t = e / (K5 * CCH);
        int rem = e % (K5 * CCH);
        int k = rem / CCH, c = rem % CCH;
        int r = tile * TM + t;
        int rr = (r < BSL) ? r : 0;
        int b = rr / L1, l = rr % L1;
        int idx = l - k * dil;
        int ic = idx < 0 ? 0 : idx;
        G[t][k][c] = encIn[(b * L1 + ic) * CCH + c];
    }
    __syncthreads();

    // --- phase 1: per-(row,h) tiny MLP + outer-product accumulate -> A2 cols 0..511 ---
    {
        int t = tid >> 4;        // 0..15
        int h = tid & 15;        // 0..15
        float acc[CCH];
#pragma unroll
        for (int c = 0; c < CCH; ++c) acc[c] = 0.f;
        for (int k = 0; k < K5; ++k) {
            float m = mv[t][k];
            if (m != 0.f) {
                float dt = dtv[t][k];
                float h2 = sk2b[h];
#pragma unroll
                for (int j = 0; j < HD; ++j) {
                    float h1 = leaky(dt * sk1W[j] + sk1b[j]);
                    h2 += h1 * sk2W[j * HD + h];
                }
                h2 = leaky(h2);                       // mask m==1 here
#pragma unroll
                for (int c = 0; c < CCH; ++c) acc[c] += G[t][k][c] * h2;
            }
        }
#pragma unroll
        for (int c = 0; c < CCH; ++c) A2[t][c * HD + h] = acc[c];
    }
    // --- phase 1b: bias-path (masked feature sums) and skip-path columns ---
    for (int e = tid; e < TM * CCH; e += 256) {
        int t = e / CCH, c = e % CCH;
        float s = 0.f;
#pragma unroll
        for (int k = 0; k < K5; ++k) s += G[t][k][c] * mv[t][k];
        A2[t][512 + c] = s;
        A2[t][544 + c] = G[t][0][c];    // tap 0 has lag 0 -> feats at l itself
    }
    __syncthreads();

    // --- phase 2: WMMA f32 16x16x4 GEMM, K split across 8 waves (72 each) ---
    {
        v8f acc0 = {};
        v8f acc1 = {};
        const v2f* __restrict__ Wp = (const v2f*)W;   // K-pair interleaved
        int rowsel = lane & 15;
        int koff = (lane >> 4) * 2;      // lanes 16..31 carry K={2,3}
        int colsel = lane & 15;
        int kbase = wv * 72;
        for (int cc = 0; cc < 18; ++cc) {
            int k0 = kbase + cc * 4 + koff;          // always even
            v2f a, b0, b1;
            a.x = A2[rowsel][k0];
            a.y = A2[rowsel][k0 + 1];
            b0 = Wp[(k0 >> 1) * CCH + colsel];        // one global_load_b64
            b1 = Wp[(k0 >> 1) * CCH + colsel + 16];   // one global_load_b64
            acc0 = __builtin_amdgcn_wmma_f32_16x16x4_f32(false, a, false, b0,
                                                         (short)0, acc0, false, false);
            acc1 = __builtin_amdgcn_wmma_f32_16x16x4_f32(false, a, false, b1,
                                                         (short)0, acc1, false, false);
        }
        int rbase = (lane >> 4) * 8;     // C/D layout: lanes 16..31 hold rows M+8
#pragma unroll
        for (int r = 0; r < 8; ++r) {
            part[wv][rbase + r][colsel] = acc0[r];
            part[wv][rbase + r][colsel + 16] = acc1[r];
        }
    }
    __syncthreads();

    // --- phase 3: reduce wave partials, add skipb, write pre-BN out, BN stats ---
    {
        int tA = wv;            // rows tA and tA+8, channel = lane
        float sA = 0.f, sB = 0.f;
#pragma unroll
        for (int w = 0; w < 8; ++w) { sA += part[w][tA][lane]; sB += part[w][tA + 8][lane]; }
        float sb = skipb[lane];
        sA += sb; sB += sb;
        int rA = tile * TM + tA;
        int rB = rA + 8;
        float cA = 0.f, cB = 0.f;
        if (rA < BSL) { outPre[rA * CCH + lane] = sA; cA = 1.f; }
        if (rB < BSL) { outPre[rB * CCH + lane] = sB; cB = 1.f; }
        float ls = sA * cA + sB * cB;
        float lq = sA * sA * cA + sB * sB * cB;
        __syncthreads();
        part[0][wv][lane] = ls;          // reuse part[] for block reduction
        part[0][wv + 8][lane] = lq;
        __syncthreads();
        if (tid < 64) {
            int which = tid >> 5, d = tid & 31;
            float v = 0.f;
#pragma unroll
            for (int w = 0; w < 8; ++w) v += part[0][which * 8 + w][d];
            atomicAdd(&stats[which * 32 + d], v);
        }
    }
}

// ---------------- BatchNorm + leaky ----------------
__global__ void k_final(const float* __restrict__ outPre, const float* __restrict__ stats,
                        const float* __restrict__ gamma, const float* __restrict__ beta,
                        float* __restrict__ dst) {
    int i = blockIdx.x * blockDim.x + threadIdx.x;   // BSL*CCH
    if (i >= BSL * CCH) return;
    int d = i & 31;
    const float n = (float)BSL;
    float mu = stats[d] / n;
    float var = stats[32 + d] / n - mu * mu;
    float x = outPre[i];
    float y = (x - mu) * rsqrtf(var + BN_EPS) * gamma[d] + beta[d];
    dst[i] = leaky(y);
}

extern "C" void kernel_launch(void* const* d_in, const int* in_sizes, int n_in,
                              void* d_out, int out_size, void* d_ws, size_t ws_size,
                              hipStream_t stream) {
    const float* ev_times = (const float*)d_in[0];
    const int*   ev_types = (const int*)d_in[1];
    const float* emb   = (const float*)d_in[2];
    const float* k1W   = (const float*)d_in[3];
    const float* k1b   = (const float*)d_in[4];
    const float* k2W   = (const float*)d_in[5];
    const float* k2b   = (const float*)d_in[6];
    const float* k3W   = (const float*)d_in[7];
    const float* k3b   = (const float*)d_in[8];
    const float* skipW = (const float*)d_in[9];
    const float* skipb = (const float*)d_in[10];
    const float* gamma = (const float*)d_in[11];
    const float* beta  = (const float*)d_in[12];

    float* ws     = (float*)d_ws;
    float* times  = ws + OFF_TIMES;
    float* encA   = ws + OFF_ENCA;
    float* encB   = ws + OFF_ENCB;
    float* outPre = ws + OFF_OUTPRE;
    float* Wst    = ws + OFF_W;
    float* stats  = ws + OFF_STATS;
    int*   bos    = (int*)(ws + OFF_BOS);

    k_init<<<1, 256, 0, stream>>>(stats, bos);
    k_maxtype<<<(BS * L0 + 255) / 256, 256, 0, stream>>>(ev_types, bos);
    k_times<<<(BSL + 255) / 256, 256, 0, stream>>>(ev_times, times);
    k_embed<<<(BSL * CCH) / 256, 256, 0, stream>>>(ev_types, emb, bos, encA);
    k_wprep<<<(4 * KTOT * CCH + 255) / 256, 256, 0, stream>>>(k3W, k3b, skipW, Wst);

    const int dils[4] = {1, 2, 4, 8};
    float* encs[2] = {encA, encB};
    const float* cur = encA;
    int nblk = (BSL + TM - 1) / TM;   // 1025
    for (int i = 0; i < 4; ++i) {
        k_layer<<<nblk, 256, 0, stream>>>(times, cur, Wst + i * KTOT * CCH,
                                          k1W + i * HD, k1b + i * HD,
                                          k2W + i * HD * HD, k2b + i * HD,
                                          skipb + i * CCH, outPre,
                                          stats + i * 64, dils[i]);
        float* nxt = (i == 3) ? (float*)d_out : encs[(i + 1) & 1];
        k_final<<<(BSL * CCH) / 256, 256, 0, stream>>>(outPre, stats + i * 64,
                                                       gamma + i * CCH, beta + i * CCH, nxt);
        cur = nxt;
    }
}